// GetImgVolume_11733850652934
// MI455X (gfx1250) — compile-verified
//
#include <hip/hip_runtime.h>
#include <stdint.h>

// ---------------- problem constants (from the reference) ----------------
constexpr int   Bn = 2, Cn = 3, Hn = 256, Wn = 320, Dn = 64;
constexpr int   HW = Hn * Wn;
constexpr int   NHALF = Dn * Hn * Wn;            // 5,242,880 = one batch of noise
constexpr float INV_MIN = 1.0f / 10.0f;          // 1/MIN_DEPTH
constexpr float INV_MAX = 1.0f / 40.0f;          // 1/MAX_DEPTH

typedef __attribute__((ext_vector_type(2))) float v2f;
typedef __attribute__((ext_vector_type(8))) float v8f;

// ---------------- JAX threefry2x32 with key = jax.random.key(1) = (0,1) ----
__device__ __forceinline__ uint32_t rotl32(uint32_t v, int n) {
  return (v << n) | (v >> (32 - n));
}

__device__ __forceinline__ void threefry2x32_k01(uint32_t& x0, uint32_t& x1) {
  const uint32_t ka = 0u, kb = 1u;
  const uint32_t kc = ka ^ kb ^ 0x1BD11BDAu;
  const int R0[4] = {13, 15, 26, 6};
  const int R1[4] = {17, 29, 16, 24};
  x0 += ka; x1 += kb;
#pragma unroll
  for (int i = 0; i < 4; i++) { x0 += x1; x1 = rotl32(x1, R0[i]); x1 ^= x0; }
  x0 += kb; x1 += kc + 1u;
#pragma unroll
  for (int i = 0; i < 4; i++) { x0 += x1; x1 = rotl32(x1, R1[i]); x1 ^= x0; }
  x0 += kc; x1 += ka + 2u;
#pragma unroll
  for (int i = 0; i < 4; i++) { x0 += x1; x1 = rotl32(x1, R0[i]); x1 ^= x0; }
  x0 += ka; x1 += kb + 3u;
#pragma unroll
  for (int i = 0; i < 4; i++) { x0 += x1; x1 = rotl32(x1, R1[i]); x1 ^= x0; }
  x0 += kb; x1 += kc + 4u;
#pragma unroll
  for (int i = 0; i < 4; i++) { x0 += x1; x1 = rotl32(x1, R0[i]); x1 ^= x0; }
  x0 += kc; x1 += ka + 5u;
}

__device__ __forceinline__ float u01(uint32_t bits) {
  // jax uniform: (bits>>9)|0x3f800000 bitcast f32, minus 1.0
  uint32_t u = (bits >> 9) | 0x3f800000u;
  return __uint_as_float(u) - 1.0f;
}

// ---------------- bilinear gather descriptor (zeros padding) ---------------
struct Gather {
  int   o00, o10, o01, o11;
  float w00, w10, w01, w11;
};

__device__ __forceinline__ Gather project_gather(const float* __restrict__ M,
                                                 float fx, float fy, float depth) {
  // M: rot[9] row-major, trans[3]
  float rx = M[0] * fx + M[1] * fy + M[2];
  float ry = M[3] * fx + M[4] * fy + M[5];
  float rz = M[6] * fx + M[7] * fy + M[8];
  float px_ = rx * depth + M[9];
  float py_ = ry * depth + M[10];
  float pz_ = rz * depth + M[11];
  bool  neg = pz_ <= 0.001f;
  float z   = neg ? 1.0f : pz_;
  float iz  = 1.0f / z;
  float px  = (neg ? (float)Wn : px_) * iz;
  float py  = (neg ? (float)Hn : py_) * iz;

  float x0f = floorf(px), y0f = floorf(py);
  float wx = px - x0f, wy = py - y0f;
  float x1f = x0f + 1.0f, y1f = y0f + 1.0f;
  float vx0 = (x0f >= 0.0f && x0f <= (float)(Wn - 1)) ? 1.0f : 0.0f;
  float vx1 = (x1f >= 0.0f && x1f <= (float)(Wn - 1)) ? 1.0f : 0.0f;
  float vy0 = (y0f >= 0.0f && y0f <= (float)(Hn - 1)) ? 1.0f : 0.0f;
  float vy1 = (y1f >= 0.0f && y1f <= (float)(Hn - 1)) ? 1.0f : 0.0f;
  int xi0 = (int)fminf(fmaxf(x0f, 0.0f), (float)(Wn - 1));
  int xi1 = (int)fminf(fmaxf(x1f, 0.0f), (float)(Wn - 1));
  int yi0 = (int)fminf(fmaxf(y0f, 0.0f), (float)(Hn - 1));
  int yi1 = (int)fminf(fmaxf(y1f, 0.0f), (float)(Hn - 1));

  Gather g;
  g.o00 = yi0 * Wn + xi0;  g.o10 = yi0 * Wn + xi1;
  g.o01 = yi1 * Wn + xi0;  g.o11 = yi1 * Wn + xi1;
  g.w00 = (1.0f - wx) * (1.0f - wy) * vx0 * vy0;
  g.w10 = wx * (1.0f - wy) * vx1 * vy0;
  g.w01 = (1.0f - wx) * wy * vx0 * vy1;
  g.w11 = wx * wy * vx1 * vy1;
  return g;
}

// ---------------- main cost-volume kernel ---------------------------------
// One thread per (d, y, x); handles b=0 and b=1 with a single threefry block
// (counter pair (i, i+N/2) is exactly (b=0, b=1) at the same (d,y,x)).
__global__ __launch_bounds__(256) void volume_kernel(
    const float* __restrict__ left, const float* __restrict__ right,
    const float* __restrict__ mats, float* __restrict__ out) {
  int gid = blockIdx.x * blockDim.x + threadIdx.x;
  if (gid >= NHALF) return;
  int x = gid % Wn;
  int t = gid / Wn;
  int y = t % Hn;
  int d = t / Hn;

  // gfx1250 prefetch hint for the image rows this thread will gather from
  __builtin_prefetch(left + y * Wn + x, 0, 1);
  __builtin_prefetch(right + y * Wn + x, 0, 1);

  uint32_t c0 = (uint32_t)gid;
  uint32_t c1 = (uint32_t)(gid + NHALF);
  threefry2x32_k01(c0, c1);

  float fx = (float)x, fy = (float)y, fd = (float)d;

#pragma unroll
  for (int b = 0; b < 2; ++b) {
    float noise = u01(b == 0 ? c0 : c1);
    float s     = noise + fd;
    float depth = 1.0f / (INV_MAX + s * (1.0f / (float)Dn) * (INV_MIN - INV_MAX));

    const float* M  = mats + b * 24;            // [rotL9,transL3, rotR9,transR3]
    Gather gl = project_gather(M,      fx, fy, depth);
    Gather gr = project_gather(M + 12, fx, fy, depth);

    const float* Lb = left  + b * (Cn * HW);
    const float* Rb = right + b * (Cn * HW);
    float acc = 0.0f;
#pragma unroll
    for (int c = 0; c < Cn; ++c) {
      const float* Lc = Lb + c * HW;
      const float* Rc = Rb + c * HW;
      float sl = gl.w00 * Lc[gl.o00] + gl.w10 * Lc[gl.o10] +
                 gl.w01 * Lc[gl.o01] + gl.w11 * Lc[gl.o11];
      float sr = gr.w00 * Rc[gr.o00] + gr.w10 * Rc[gr.o10] +
                 gr.w01 * Rc[gr.o01] + gr.w11 * Rc[gr.o11];
      acc += fabsf(sr - sl);
    }
    out[((b * Dn + d) * Hn + y) * Wn + x] = acc;
  }
}

// ---------------- setup kernel 1: 4x4 inverse of gt_proj per batch ---------
__global__ void invert4x4_kernel(const float* __restrict__ gt,
                                 float* __restrict__ invout, int nb) {
  int b = blockIdx.x * blockDim.x + threadIdx.x;
  if (b >= nb) return;
  float a[4][4], v[4][4];
  for (int i = 0; i < 4; i++)
    for (int j = 0; j < 4; j++) {
      a[i][j] = gt[b * 16 + i * 4 + j];
      v[i][j] = (i == j) ? 1.0f : 0.0f;
    }
  for (int col = 0; col < 4; ++col) {
    int p = col; float best = fabsf(a[col][col]);
    for (int r = col + 1; r < 4; ++r) {
      float t = fabsf(a[r][col]);
      if (t > best) { best = t; p = r; }
    }
    if (p != col)
      for (int j = 0; j < 4; j++) {
        float t = a[col][j]; a[col][j] = a[p][j]; a[p][j] = t;
        t = v[col][j]; v[col][j] = v[p][j]; v[p][j] = t;
      }
    float pv = 1.0f / a[col][col];
    for (int j = 0; j < 4; j++) { a[col][j] *= pv; v[col][j] *= pv; }
    for (int r = 0; r < 4; ++r)
      if (r != col) {
        float f = a[r][col];
        for (int j = 0; j < 4; j++) { a[r][j] -= f * a[col][j]; v[r][j] -= f * v[col][j]; }
      }
  }
  for (int i = 0; i < 4; i++)
    for (int j = 0; j < 4; j++) invout[b * 16 + i * 4 + j] = v[i][j];
}

// ---------------- setup kernel 2: proj = src_proj @ inv(gt_proj) via WMMA --
// One wave (32 threads, EXEC all ones). The 4x4 products live in the top-left
// corner of a 16x16x4 f32 WMMA. Output packed as rot[9]+trans[3] per (b,img).
__global__ __launch_bounds__(32) void proj_wmma_kernel(
    const float* __restrict__ left_proj, const float* __restrict__ right_proj,
    const float* __restrict__ invgt, float* __restrict__ mats) {
  int lane = threadIdx.x;      // 0..31
  int m    = lane & 15;
  int hi   = lane >> 4;        // 0: K=0/1 half, 1: K=2/3 half

#if defined(__gfx1250__) && __has_builtin(__builtin_amdgcn_wmma_f32_16x16x4_f32)
  for (int combo = 0; combo < 4; ++combo) {
    int b = combo >> 1, img = combo & 1;
    const float* src = (img == 0 ? left_proj : right_proj) + b * 16;
    const float* inv = invgt + b * 16;
    // A (16x4, M in lanes, K split across VGPR x lane-half): rows 0..3 = src_proj
    v2f A, Bm;
    A[0]  = (m < 4) ? src[m * 4 + (hi ? 2 : 0)] : 0.0f;
    A[1]  = (m < 4) ? src[m * 4 + (hi ? 3 : 1)] : 0.0f;
    // B (4x16, N in lanes, K split across VGPR x lane-half): cols 0..3 = inv(gt)
    Bm[0] = (m < 4) ? inv[(hi ? 2 : 0) * 4 + m] : 0.0f;
    Bm[1] = (m < 4) ? inv[(hi ? 3 : 1) * 4 + m] : 0.0f;
    v8f Cz = {};
    v8f Dr = __builtin_amdgcn_wmma_f32_16x16x4_f32(
        false, A, false, Bm, (short)0, Cz, false, false);
    // D: lanes 0..15 carry N; VGPR r carries row M=r. Lane n<4 holds P[0..3][n].
    float* outp = mats + b * 24 + img * 12;
    if (lane < 4) {
      float p0 = Dr[0], p1 = Dr[1], p2 = Dr[2];
      if (lane < 3) {
        outp[0 * 3 + lane] = p0;  // rot row-major
        outp[1 * 3 + lane] = p1;
        outp[2 * 3 + lane] = p2;
      } else {
        outp[9]  = p0;            // trans
        outp[10] = p1;
        outp[11] = p2;
      }
    }
  }
#else
  // scalar fallback (also used for the host pass)
  if (lane < 4) {
    int b = lane >> 1, img = lane & 1;
    const float* src = (img == 0 ? left_proj : right_proj) + b * 16;
    const float* inv = invgt + b * 16;
    float P[4][4];
    for (int i = 0; i < 4; i++)
      for (int j = 0; j < 4; j++) {
        float acc = 0.0f;
        for (int k = 0; k < 4; k++) acc += src[i * 4 + k] * inv[k * 4 + j];
        P[i][j] = acc;
      }
    float* outp = mats + b * 24 + img * 12;
    for (int i = 0; i < 3; i++)
      for (int j = 0; j < 3; j++) outp[i * 3 + j] = P[i][j];
    outp[9] = P[0][3]; outp[10] = P[1][3]; outp[11] = P[2][3];
  }
#endif
}

// ---------------- launcher -------------------------------------------------
extern "C" void kernel_launch(void* const* d_in, const int* in_sizes, int n_in,
                              void* d_out, int out_size, void* d_ws, size_t ws_size,
                              hipStream_t stream) {
  (void)in_sizes; (void)n_in; (void)out_size; (void)ws_size;
  const float* left  = (const float*)d_in[0];
  const float* right = (const float*)d_in[1];
  const float* lproj = (const float*)d_in[2];
  const float* rproj = (const float*)d_in[3];
  const float* gproj = (const float*)d_in[4];
  float* out = (float*)d_out;
  float* wsf = (float*)d_ws;

  float* invgt = wsf;        // B*16 floats
  float* mats  = wsf + 64;   // B*24 floats: per batch [rotL9,transL3,rotR9,transR3]

  invert4x4_kernel<<<1, Bn, 0, stream>>>(gproj, invgt, Bn);
  proj_wmma_kernel<<<1, 32, 0, stream>>>(lproj, rproj, invgt, mats);

  const int threads = 256;
  const int blocks  = (NHALF + threads - 1) / threads;   // 20480
  volume_kernel<<<blocks, threads, 0, stream>>>(left, right, mats, out);
}